// FastFlexAddAttention_41248865911339
// MI455X (gfx1250) — compile-verified
//
#include <hip/hip_runtime.h>
#include <math.h>

// Problem constants from the reference: T=131072, C=512, O=512, N=1024
#define C_DIM 512
#define O_DIM 512
#define N_SEG 1024

typedef __attribute__((ext_vector_type(2))) float v2f;
typedef __attribute__((ext_vector_type(8))) float v8f;

// Monotone order-preserving float<->uint encoding so we can use integer atomicMax.
__device__ __forceinline__ unsigned f2ord(float f) {
  unsigned u = __float_as_uint(f);
  return (u & 0x80000000u) ? ~u : (u | 0x80000000u);
}
__device__ __forceinline__ float ord2f(unsigned u) {
  return (u & 0x80000000u) ? __uint_as_float(u & 0x7fffffffu)
                           : __uint_as_float(~u);
}

// ---------------------------------------------------------------------------
// Kernel 0: zero y[N,C] and denom[N]; seg_max = encoded -inf.
// ---------------------------------------------------------------------------
__global__ void k_init(float* __restrict__ y, float* __restrict__ denom,
                       unsigned* __restrict__ segmax_u) {
  int i = blockIdx.x * blockDim.x + threadIdx.x;
  if (i < N_SEG * C_DIM) y[i] = 0.0f;
  if (i < N_SEG) {
    denom[i] = 0.0f;
    segmax_u[i] = 0x007FFFFFu;  // f2ord(-inf)
  }
}

// ---------------------------------------------------------------------------
// Kernel 1: score[t] = x[t]·ws + bs, and per-segment running max via atomicMax.
// One wave32 per row; lane L loads float4 at column 4L + 128j (perfectly
// coalesced 512B per wave per load). ws is loop-invariant -> cached in regs.
// ---------------------------------------------------------------------------
__global__ void k_score(const float* __restrict__ x, const float* __restrict__ ws,
                        const float* __restrict__ bs, const int* __restrict__ seg,
                        float* __restrict__ score, unsigned* __restrict__ segmax_u,
                        int T) {
  const int lane = threadIdx.x & 31;
  const int wave = threadIdx.x >> 5;  // 0..7

  float4 wsv[4];
#pragma unroll
  for (int j = 0; j < 4; ++j)
    wsv[j] = *(const float4*)(ws + j * 128 + lane * 4);
  const float bias = bs[0];

  const int rowbase = blockIdx.x * 64;
#pragma unroll 1
  for (int i = 0; i < 8; ++i) {
    const int row = rowbase + i * 8 + wave;
    if (row >= T) break;
    const float* xr = x + (size_t)row * C_DIM;
    float acc = 0.0f;
#pragma unroll
    for (int j = 0; j < 4; ++j) {
      float4 xv = *(const float4*)(xr + j * 128 + lane * 4);
      acc = fmaf(xv.x, wsv[j].x, acc);
      acc = fmaf(xv.y, wsv[j].y, acc);
      acc = fmaf(xv.z, wsv[j].z, acc);
      acc = fmaf(xv.w, wsv[j].w, acc);
    }
#pragma unroll
    for (int off = 16; off > 0; off >>= 1)
      acc += __shfl_xor(acc, off, 32);
    if (lane == 0) {
      const float s = acc + bias;
      score[row] = s;
      atomicMax(&segmax_u[seg[row]], f2ord(s));
    }
  }
}

// ---------------------------------------------------------------------------
// Kernel 2: y[seg] += e_t * x[t], denom[seg] += e_t, with e_t = exp(s - max).
// One block per 128 sorted rows; thread owns a float2 column slice and keeps a
// register accumulator while the segment id is unchanged (segment ids are
// sorted -> ~1 boundary flush per block via f32 global atomics).
// ---------------------------------------------------------------------------
__global__ void k_accum(const float* __restrict__ x, const float* __restrict__ score,
                        const unsigned* __restrict__ segmax_u,
                        const int* __restrict__ seg,
                        float* __restrict__ denom, float* __restrict__ y, int T) {
  __shared__ float eArr[128];
  __shared__ int sArr[128];
  const int rowbase = blockIdx.x * 128;
  const int tid = threadIdx.x;

  if (tid < 128) {
    const int row = rowbase + tid;
    const int s = seg[row];
    const float m = ord2f(segmax_u[s]);
    const float e = expf(score[row] - m);
    sArr[tid] = s;
    eArr[tid] = e;
    atomicAdd(&denom[s], e);
  }
  __syncthreads();

  const int col = tid * 2;  // 256 threads x 2 cols = 512
  float ax = 0.0f, ay = 0.0f;
  int cur = sArr[0];
  const float* xp = x + (size_t)rowbase * C_DIM + col;
#pragma unroll 1
  for (int r = 0; r < 128; ++r) {
    const int s = sArr[r];
    if (s != cur) {
      atomicAdd(&y[(size_t)cur * C_DIM + col], ax);
      atomicAdd(&y[(size_t)cur * C_DIM + col + 1], ay);
      ax = 0.0f;
      ay = 0.0f;
      cur = s;
    }
    __builtin_prefetch(xp + 8 * C_DIM, 0, 0);  // global_prefetch_b8, 16KB (8 rows) ahead
    const float2 xv = *(const float2*)xp;
    const float e = eArr[r];
    ax = fmaf(e, xv.x, ax);
    ay = fmaf(e, xv.y, ay);
    xp += C_DIM;
  }
  atomicAdd(&y[(size_t)cur * C_DIM + col], ax);
  atomicAdd(&y[(size_t)cur * C_DIM + col + 1], ay);
}

// ---------------------------------------------------------------------------
// Kernel 3: out[N,O] = (y @ Wp)/denom + bp (0 for empty segments) using
// V_WMMA_F32_16X16X4_F32. One wave per 16x16 output tile, K stepped by 4.
// A (16x4 f32): lanes 0-15 hold K=0,1 in v[0],v[1]; lanes 16-31 hold K=2,3.
// B (4x16 f32): mirrored layout. C/D: VGPR v -> rows v (lanes 0-15) / v+8.
// ---------------------------------------------------------------------------
__global__ void k_gemm(const float* __restrict__ y, const float* __restrict__ Wp,
                       const float* __restrict__ bp, const float* __restrict__ denom,
                       float* __restrict__ out) {
  const int lane = threadIdx.x & 31;
  const int wave = threadIdx.x >> 5;
  const int tiles_n = O_DIM / 16;  // 32
  const int gt = blockIdx.x * (blockDim.x >> 5) + wave;
  const int tm = gt / tiles_n;
  const int tn = gt - tm * tiles_n;
  if (tm >= N_SEG / 16) return;

  const int l15 = lane & 15;
  const int koff = (lane >> 4) * 2;  // half-wave K offset
  const float* arow = y + (size_t)(tm * 16 + l15) * C_DIM + koff;
  const float* bcol = Wp + (size_t)koff * O_DIM + tn * 16 + l15;

  v8f acc = {};
#pragma unroll 4
  for (int k = 0; k < C_DIM; k += 4) {
    v2f a, b;
    a.x = arow[0];
    a.y = arow[1];
    b.x = bcol[0];
    b.y = bcol[O_DIM];
    acc = __builtin_amdgcn_wmma_f32_16x16x4_f32(
        /*neg_a=*/false, a, /*neg_b=*/false, b,
        /*c_mod=*/(short)0, acc, /*reuse_a=*/false, /*reuse_b=*/false);
    arow += 4;
    bcol += 4 * O_DIM;
  }

  const int n = tn * 16 + l15;
  const float bias = bp[n];
  const int mbase = tm * 16 + ((lane >> 4) * 8);
#pragma unroll
  for (int v = 0; v < 8; ++v) {
    const int m = mbase + v;
    const float den = denom[m];
    const float val = (den > 0.0f) ? (acc[v] / den + bias) : 0.0f;
    out[(size_t)m * O_DIM + n] = val;
  }
}

// ---------------------------------------------------------------------------
extern "C" void kernel_launch(void* const* d_in, const int* in_sizes, int n_in,
                              void* d_out, int out_size, void* d_ws, size_t ws_size,
                              hipStream_t stream) {
  const float* x  = (const float*)d_in[0];
  const float* Wp = (const float*)d_in[1];
  const float* bp = (const float*)d_in[2];
  const float* ws = (const float*)d_in[3];
  const float* bs = (const float*)d_in[4];
  const int* seg  = (const int*)d_in[5];  // harness: integer -> const int*
  const int T = in_sizes[5];
  float* out = (float*)d_out;

  // Workspace layout: score[T] | segmax_u[N] | denom[N] | y[N*C]  (~2.6 MB)
  char* p = (char*)d_ws;
  float* score = (float*)p;        p += (size_t)T * sizeof(float);
  unsigned* segmax = (unsigned*)p; p += (size_t)N_SEG * sizeof(unsigned);
  float* denom = (float*)p;        p += (size_t)N_SEG * sizeof(float);
  float* y = (float*)p;

  k_init<<<(N_SEG * C_DIM + 255) / 256, 256, 0, stream>>>(y, denom, segmax);
  k_score<<<(T + 63) / 64, 256, 0, stream>>>(x, ws, bs, seg, score, segmax, T);
  k_accum<<<T / 128, 256, 0, stream>>>(x, score, segmax, seg, denom, y, T);
  k_gemm<<<(N_SEG / 16) * (O_DIM / 16) / 8, 256, 0, stream>>>(y, Wp, bp, denom, out);
}